// SemanticMambaLayer_52080773431747
// MI455X (gfx1250) — compile-verified
//
#include <hip/hip_runtime.h>
#include <cstddef>

// ---------------- model constants ----------------
#define NB      2      // batch
#define CDIM    96     // DIM
#define CHID    192    // HID = 2*DIM
#define NSEQ    8192   // tokens per batch (8*32*32)
#define NTOT    16384  // NB*NSEQ
#define NEXP    64     // NT
#define DSTATE  16     // DST
#define IRANK   32     // IR
#define NHEAD   6      // NH
#define RKD     6      // RK = DIM/16
#define LDXD    22     // RKD + DSTATE
#define WINS    16     // WS
#define NDEPTH  4
#define NCH     64     // scan chunks per batch
#define CLEN    128    // NSEQ / NCH
#define LOG2E   1.4426950408889634f

typedef float v2f __attribute__((ext_vector_type(2)));
typedef float v8f __attribute__((ext_vector_type(8)));

__device__ __forceinline__ float gelu_f(float x) {
  float x3 = x * x * x;
  return 0.5f * x * (1.0f + tanhf(0.7978845608028654f * (x + 0.044715f * x3)));
}
__device__ __forceinline__ float softplus_f(float x) {
  return (x > 20.0f) ? x : log1pf(__expf(x));
}

// DPP16 ROW_XMASK reductions: xor within 16-lane rows, pure VALU (no LDS).
template <int XM>
__device__ __forceinline__ float row_xadd(float x) {
  int y = __builtin_amdgcn_mov_dpp(__float_as_int(x), 0x160 | XM, 0xF, 0xF, true);
  return x + __int_as_float(y);
}
template <int XM>
__device__ __forceinline__ float row_xmax(float x) {
  int y = __builtin_amdgcn_mov_dpp(__float_as_int(x), 0x160 | XM, 0xF, 0xF, true);
  return fmaxf(x, __int_as_float(y));
}
__device__ __forceinline__ float red16_add(float x) {
  x = row_xadd<1>(x); x = row_xadd<2>(x); x = row_xadd<4>(x); return row_xadd<8>(x);
}
__device__ __forceinline__ float red16_max(float x) {
  x = row_xmax<1>(x); x = row_xmax<2>(x); x = row_xmax<4>(x); return row_xmax<8>(x);
}

// ---------------- LN over channel dim, input in (B,C,N) layout ----------------
__global__ void ln_in_kernel(const float* __restrict__ x, const float* __restrict__ w,
                             const float* __restrict__ b, float* __restrict__ h) {
  int t = blockIdx.x * blockDim.x + threadIdx.x;  // token id, < NTOT
  int bb = t / NSEQ, n = t % NSEQ;
  const float* px = x + (size_t)bb * CDIM * NSEQ + n;
  float s = 0.0f;
  for (int c = 0; c < CDIM; ++c) s += px[(size_t)c * NSEQ];
  float m = s * (1.0f / CDIM);
  float v = 0.0f;
  for (int c = 0; c < CDIM; ++c) { float d = px[(size_t)c * NSEQ] - m; v += d * d; }
  float rs = __builtin_amdgcn_rsqf(v * (1.0f / CDIM) + 1e-5f);
  float* ph = h + (size_t)t * CDIM;
  for (int c = 0; c < CDIM; ++c) ph[c] = (px[(size_t)c * NSEQ] - m) * rs * w[c] + b[c];
}

// ---------------- LN over rows (token-major), wave per row ----------------
__global__ void ln_rows_kernel(const float* __restrict__ in, const float* __restrict__ w,
                               const float* __restrict__ b, float* __restrict__ out, int C) {
  int row = blockIdx.x * (blockDim.x >> 5) + (threadIdx.x >> 5);
  int lane = threadIdx.x & 31;
  const float* pr = in + (size_t)row * C;
  float s = 0.0f;
  for (int c = lane; c < C; c += 32) s += pr[c];
  for (int o = 16; o > 0; o >>= 1) s += __shfl_xor(s, o, 32);
  float m = s / (float)C;
  float v = 0.0f;
  for (int c = lane; c < C; c += 32) { float d = pr[c] - m; v += d * d; }
  for (int o = 16; o > 0; o >>= 1) v += __shfl_xor(v, o, 32);
  float rs = __builtin_amdgcn_rsqf(v / (float)C + 1e-5f);
  float* po = out + (size_t)row * C;
  for (int c = lane; c < C; c += 32) po[c] = (pr[c] - m) * rs * w[c] + b[c];
}

// ---------------- fully-specialized WMMA f32 GEMM ----------------
template <int K, int LDA, int LDW, int LDC, int NC, int ACT, bool ACC, int MT>
__global__ void gemm16_wmma(const float* __restrict__ A, const float* __restrict__ W,
                            const float* __restrict__ bias, float* __restrict__ C) {
  constexpr bool CG = (NC % 16) != 0;     // column guard needed (NC=22 case)
  constexpr int KSTEPS = (K + 3) / 4;
  int lane = threadIdx.x;
  int half = lane >> 4, lc = lane & 15;
  int col = blockIdx.y * 16 + lc;
  bool cok = true;
  int ccl = col;
  if constexpr (CG) { cok = col < NC; ccl = cok ? col : 0; }
  const float* arow0 = A + (size_t)(blockIdx.x * (16 * MT) + lc) * LDA;
  const float* arow1 = arow0 + (size_t)16 * LDA;
  v8f acc0 = {}, acc1 = {};
#pragma unroll
  for (int ks = 0; ks < KSTEPS; ++ks) {
    int k0 = ks * 4;
    int ka = k0 + 2 * half;
    bool kok = ((K & 3) == 0) || (ka + 1 < K);   // compile-time true when K%4==0
    int kc = kok ? ka : 0;                       // clamp -> unconditional loads
    v2f a0 = *(const v2f*)(arow0 + kc);
    v2f a1 = {};
    if constexpr (MT == 2) a1 = *(const v2f*)(arow1 + kc);
    v2f b;
    b.x = W[(size_t)kc * LDW + ccl];
    b.y = W[(size_t)(kc + 1) * LDW + ccl];
    float msk = 1.0f;
    if ((K & 3) != 0) msk = kok ? 1.0f : 0.0f;
    if constexpr (CG) msk = cok ? msk : 0.0f;
    if ((K & 3) != 0 || CG) b *= msk;            // branchless zeroing
    acc0 = __builtin_amdgcn_wmma_f32_16x16x4_f32(false, a0, false, b, (short)0, acc0,
                                                 false, false);
    if constexpr (MT == 2)
      acc1 = __builtin_amdgcn_wmma_f32_16x16x4_f32(false, a1, false, b, (short)0, acc1,
                                                   false, false);
  }
  if constexpr (CG) { if (!cok) return; }
  float bval = bias ? bias[col] : 0.0f;
  int m0 = blockIdx.x * (16 * MT) + half * 8;
#pragma unroll
  for (int r = 0; r < 8; ++r) {
    float v = acc0[r] + bval;
    if constexpr (ACT == 1) v = gelu_f(v);
    if constexpr (ACT == 2) v = softplus_f(v);
    size_t o = (size_t)(m0 + r) * LDC + col;
    if constexpr (ACC) C[o] += v; else C[o] = v;
  }
  if constexpr (MT == 2) {
#pragma unroll
    for (int r = 0; r < 8; ++r) {
      float v = acc1[r] + bval;
      if constexpr (ACT == 1) v = gelu_f(v);
      if constexpr (ACT == 2) v = softplus_f(v);
      size_t o = (size_t)(m0 + 16 + r) * LDC + col;
      if constexpr (ACC) C[o] += v; else C[o] = v;
    }
  }
}

template <int K, int LDA, int LDW, int LDC, int NC, int ACT, bool ACC, int MT>
static inline void gemm_launch(hipStream_t st, const float* A, const float* W,
                               const float* bias, float* C, int M) {
  dim3 g(M / (16 * MT), (NC + 15) / 16);
  gemm16_wmma<K, LDA, LDW, LDC, NC, ACT, ACC, MT><<<g, 32, 0, st>>>(A, W, bias, C);
}

// ---------------- softmax over NEXP=64 + stable argmax + histogram ----------------
__global__ void softmax_argmax_kernel(float* __restrict__ L, int* __restrict__ key,
                                      int* __restrict__ hist) {
  int t = blockIdx.x * (blockDim.x >> 5) + (threadIdx.x >> 5);
  int lane = threadIdx.x & 31;
  float* row = L + (size_t)t * NEXP;
  float v0 = row[lane], v1 = row[lane + 32];
  float mv; int mi;
  if (v1 > v0) { mv = v1; mi = lane + 32; } else { mv = v0; mi = lane; }
  for (int o = 16; o > 0; o >>= 1) {
    float ov = __shfl_xor(mv, o, 32);
    int oi = __shfl_xor(mi, o, 32);
    if (ov > mv || (ov == mv && oi < mi)) { mv = ov; mi = oi; }
  }
  float e0 = __expf(v0 - mv), e1 = __expf(v1 - mv);
  float s = e0 + e1;
  for (int o = 16; o > 0; o >>= 1) s += __shfl_xor(s, o, 32);
  float inv = __builtin_amdgcn_rcpf(s);
  row[lane] = e0 * inv;
  row[lane + 32] = e1 * inv;
  if (lane == 0) {
    key[t] = mi;
    atomicAdd(&hist[(t / NSEQ) * NEXP + mi], 1);
  }
}

// ---------------- stable counting sort per batch (64 buckets) ----------------
__global__ void sort_kernel(const int* __restrict__ key, const int* __restrict__ hist,
                            int* __restrict__ order, int* __restrict__ inv) {
  int gb = blockIdx.x;
  __shared__ int base[NEXP];
  if (threadIdx.x == 0) {
    int s = 0;
    for (int j = 0; j < NEXP; ++j) { base[j] = s; s += hist[gb * NEXP + j]; }
  }
  __syncthreads();
  int k = threadIdx.x;
  int pos = base[k];
  const int* kb = key + (size_t)gb * NSEQ;
  for (int n = 0; n < NSEQ; ++n) {
    if (kb[n] == k) {
      order[(size_t)gb * NSEQ + pos] = n;
      inv[(size_t)gb * NSEQ + n] = pos;
      ++pos;
    }
  }
}

// ---------------- gather rows by order: hs/ps ----------------
__global__ void gather_kernel(const float* __restrict__ HI, const float* __restrict__ PR,
                              const int* __restrict__ order, float* __restrict__ HS,
                              float* __restrict__ PS) {
  int t = blockIdx.x;
  int gb = t / NSEQ;
  int src = gb * NSEQ + order[t];
  for (int c = threadIdx.x; c < CHID; c += blockDim.x)
    HS[(size_t)t * CHID + c] = HI[(size_t)src * CHID + c];
  if (threadIdx.x < DSTATE)
    PS[(size_t)t * DSTATE + threadIdx.x] = PR[(size_t)src * DSTATE + threadIdx.x];
}

// ================= chunked parallel selective scan =================
// Recurrence h_t = exp2(dt_t*Al2)*h_{t-1} + dt_t*u_t*b_t is linear, so:
// pass1: per-chunk end state from 0 + per-chunk sum(dt)  (chunk A-product =
//        exp2(Al2 * sum dt)); combine: sequential fold of 64 chunk summaries
//        into per-chunk initial states; pass2: replay chunk from true initial
//        state and emit y. Serial critical path: 8192 -> ~128+64+128.
__global__ void scan_pass1(const float* __restrict__ u, const float* __restrict__ dt,
                           const float* __restrict__ XDm, const float* __restrict__ A_log,
                           float* __restrict__ HEND, float* __restrict__ SUMDT) {
  int blk = blockIdx.x;
  int chunk = blk % NCH;
  int chgrp = (blk / NCH) % (CHID / 16);
  int gb = blk / (NCH * (CHID / 16));
  int ch = chgrp * 16 + (threadIdx.x >> 4);
  int s = threadIdx.x & 15;
  float Al2 = -__expf(A_log[(size_t)ch * DSTATE + s]) * LOG2E;
  float h = 0.0f, sd = 0.0f;
  size_t rbase = (size_t)gb * NSEQ + (size_t)chunk * CLEN;
#pragma unroll 4
  for (int t = 0; t < CLEN; ++t) {
    size_t r = rbase + t;
    float dtv = dt[r * CHID + ch];
    float uv  = u[r * CHID + ch];
    float bv  = XDm[r * LDXD + RKD + s];
    h = __builtin_amdgcn_exp2f(dtv * Al2) * h + dtv * uv * bv;
    sd += dtv;
  }
  size_t st = (size_t)(gb * CHID + ch) * DSTATE + s;
  HEND[st * NCH + chunk] = h;
  if (s == 0) SUMDT[(size_t)(gb * CHID + ch) * NCH + chunk] = sd;
}

__global__ void scan_combine(const float* __restrict__ A_log,
                             const float* __restrict__ HEND,
                             const float* __restrict__ SUMDT,
                             float* __restrict__ HINIT) {
  int blk = blockIdx.x;  // NB * CHID/16
  int chgrp = blk % (CHID / 16);
  int gb = blk / (CHID / 16);
  int ch = chgrp * 16 + (threadIdx.x >> 4);
  int s = threadIdx.x & 15;
  float Al2 = -__expf(A_log[(size_t)ch * DSTATE + s]) * LOG2E;
  size_t st = (size_t)(gb * CHID + ch) * DSTATE + s;
  size_t cb = (size_t)(gb * CHID + ch) * NCH;
  float run = 0.0f;
  for (int c = 0; c < NCH; ++c) {
    HINIT[st * NCH + c] = run;
    run = __builtin_amdgcn_exp2f(Al2 * SUMDT[cb + c]) * run + HEND[st * NCH + c];
  }
}

__global__ void scan_pass2(const float* __restrict__ u, const float* __restrict__ dt,
                           const float* __restrict__ XDm, const float* __restrict__ Cm,
                           const float* __restrict__ A_log, const float* __restrict__ Dk,
                           const float* __restrict__ HINIT, float* __restrict__ y) {
  int blk = blockIdx.x;
  int chunk = blk % NCH;
  int chgrp = (blk / NCH) % (CHID / 16);
  int gb = blk / (NCH * (CHID / 16));
  int ch = chgrp * 16 + (threadIdx.x >> 4);
  int s = threadIdx.x & 15;
  float Al2 = -__expf(A_log[(size_t)ch * DSTATE + s]) * LOG2E;
  float Dv = Dk[ch];
  size_t st = (size_t)(gb * CHID + ch) * DSTATE + s;
  float h = HINIT[st * NCH + chunk];
  size_t rbase = (size_t)gb * NSEQ + (size_t)chunk * CLEN;
#pragma unroll 4
  for (int t = 0; t < CLEN; ++t) {
    size_t r = rbase + t;
    float dtv = dt[r * CHID + ch];
    float uv  = u[r * CHID + ch];
    float bv  = XDm[r * LDXD + RKD + s];
    float cv  = Cm[r * DSTATE + s];
    h = __builtin_amdgcn_exp2f(dtv * Al2) * h + dtv * uv * bv;
    float p = red16_add(h * cv);        // DPP row_xmask reduction over states
    if (s == 0) y[r * CHID + ch] = p + uv * Dv;
  }
}

// ---------------- scatter-add: h[t] += Z[inv[t]] ----------------
__global__ void scatter_add_kernel(const float* __restrict__ Z, const int* __restrict__ inv,
                                   float* __restrict__ h) {
  int idx = blockIdx.x * blockDim.x + threadIdx.x;  // < NTOT*CDIM
  int t = idx / CDIM, c = idx % CDIM;
  int gb = t / NSEQ;
  h[idx] += Z[((size_t)gb * NSEQ + inv[t]) * CDIM + c];
}

// ---------------- window attention: wave = (batch, window, head), WMMA 16x16 ----------------
__global__ void winattn_kernel(const float* __restrict__ qkv, float* __restrict__ out) {
  __shared__ float lds[8 * 256];
  int wid = threadIdx.x >> 5, lane = threadIdx.x & 31;
  int half = lane >> 4, lc = lane & 15;
  int task = blockIdx.x * 8 + wid;
  int head = task % NHEAD;
  int win = (task / NHEAD) % (NSEQ / WINS);
  int gb = task / (NHEAD * (NSEQ / WINS));
  size_t base = ((size_t)gb * NSEQ + (size_t)win * WINS) * (3 * CDIM);
  int qo = head * 16, ko = CDIM + head * 16, vo = 2 * CDIM + head * 16;

  v8f acc = {};
#pragma unroll
  for (int k0 = 0; k0 < 16; k0 += 4) {   // S = q @ k^T
    int ka = k0 + 2 * half;
    v2f av = *(const v2f*)(&qkv[base + (size_t)lc * (3 * CDIM) + qo + ka]);
    v2f bv = *(const v2f*)(&qkv[base + (size_t)lc * (3 * CDIM) + ko + ka]);
    acc = __builtin_amdgcn_wmma_f32_16x16x4_f32(false, av, false, bv, (short)0, acc,
                                                false, false);
  }
  float* P = &lds[wid * 256];
#pragma unroll
  for (int r = 0; r < 8; ++r) {          // row softmax within 16-lane halves (DPP)
    float v = acc[r] * 0.25f;            // hd^-0.5
    float mx = red16_max(v);
    float e = __expf(v - mx);
    float sm = red16_add(e);
    P[(r + 8 * half) * 16 + lc] = e * __builtin_amdgcn_rcpf(sm);
  }
  v8f o2 = {};
#pragma unroll
  for (int k0 = 0; k0 < 16; k0 += 4) {   // O = P @ v
    int ka = k0 + 2 * half;
    v2f av = *(const v2f*)(&P[lc * 16 + ka]);
    v2f bv;
    bv.x = qkv[base + (size_t)ka * (3 * CDIM) + vo + lc];
    bv.y = qkv[base + (size_t)(ka + 1) * (3 * CDIM) + vo + lc];
    o2 = __builtin_amdgcn_wmma_f32_16x16x4_f32(false, av, false, bv, (short)0, o2,
                                               false, false);
  }
  size_t trow = (size_t)gb * NSEQ + (size_t)win * WINS;
#pragma unroll
  for (int r = 0; r < 8; ++r)
    out[(trow + r + 8 * half) * CDIM + head * 16 + lc] = o2[r];
}

// ---------------- depthwise 5x5x5 conv + residual, channel-fastest ----------------
__global__ void dwconv_kernel(const float* __restrict__ F1, const float* __restrict__ w,
                              float* __restrict__ F2) {
  int idx = blockIdx.x * blockDim.x + threadIdx.x;  // < NTOT*CHID
  int c = idx % CHID;
  int t = idx / CHID;
  int n = t % NSEQ;
  int gb = t / NSEQ;
  int d = n >> 10, hh = (n >> 5) & 31, wx = n & 31;
  const float* wc = w + (size_t)c * 125;
  float acc = 0.0f;
  for (int kd = 0; kd < 5; ++kd) {
    int dd = d + kd - 2;
    if ((unsigned)dd >= 8u) continue;
    for (int kh = 0; kh < 5; ++kh) {
      int h2 = hh + kh - 2;
      if ((unsigned)h2 >= 32u) continue;
      for (int kw = 0; kw < 5; ++kw) {
        int w2 = wx + kw - 2;
        if ((unsigned)w2 >= 32u) continue;
        int nn = (dd << 10) | (h2 << 5) | w2;
        acc += F1[((size_t)gb * NSEQ + nn) * CHID + c] * wc[kd * 25 + kh * 5 + kw];
      }
    }
  }
  F2[(size_t)t * CHID + c] = F1[(size_t)t * CHID + c] + acc;
}

// ---------------- final: out(B,C,N) = h(B,N,C) + x ----------------
__global__ void out_add_kernel(const float* __restrict__ h, const float* __restrict__ x,
                               float* __restrict__ out) {
  int idx = blockIdx.x * blockDim.x + threadIdx.x;  // < NB*CDIM*NSEQ
  int n = idx % NSEQ;
  int c = (idx / NSEQ) % CDIM;
  int gb = idx / (NSEQ * CDIM);
  out[idx] = x[idx] + h[((size_t)gb * NSEQ + n) * CDIM + c];
}

// ---------------- host side ----------------
extern "C" void kernel_launch(void* const* d_in, const int* in_sizes, int n_in,
                              void* d_out, int out_size, void* d_ws, size_t ws_size,
                              hipStream_t stream) {
  (void)in_sizes; (void)n_in; (void)out_size; (void)ws_size;
  const float* x       = (const float*)d_in[0];
  const float* ln_w    = (const float*)d_in[1];
  const float* ln_b    = (const float*)d_in[2];
  const float* tokenA  = (const float*)d_in[3];
  const float* embB    = (const float*)d_in[4];
  const float* rw1     = (const float*)d_in[5];
  const float* rb1     = (const float*)d_in[6];
  const float* rw2     = (const float*)d_in[7];
  const float* rb2     = (const float*)d_in[8];
  const float* in_w    = (const float*)d_in[9];
  const float* in_b    = (const float*)d_in[10];
  const float* xproj_w = (const float*)d_in[11];
  const float* dt_w    = (const float*)d_in[12];
  const float* dt_b    = (const float*)d_in[13];
  const float* A_log   = (const float*)d_in[14];
  const float* Dskip   = (const float*)d_in[15];
  const float* onorm_w = (const float*)d_in[16];
  const float* onorm_b = (const float*)d_in[17];
  const float* out_w   = (const float*)d_in[18];
  const float* out_b   = (const float*)d_in[19];
  const float* qkv_w   = (const float*)d_in[20];
  const float* qkv_b   = (const float*)d_in[21];
  const float* proj_w  = (const float*)d_in[22];
  const float* proj_b  = (const float*)d_in[23];
  const float* n1_w    = (const float*)d_in[24];
  const float* n1_b    = (const float*)d_in[25];
  const float* n2_w    = (const float*)d_in[26];
  const float* n2_b    = (const float*)d_in[27];
  const float* fc1_w   = (const float*)d_in[28];
  const float* fc1_b   = (const float*)d_in[29];
  const float* dw_w    = (const float*)d_in[30];
  const float* fc2_w   = (const float*)d_in[31];
  const float* fc2_b   = (const float*)d_in[32];

  // workspace layout (floats)
  float* ws = (float*)d_ws;
  size_t o = 0;
  float* H    = ws + o; o += (size_t)NTOT * CDIM;      // residual stream
  float* R1   = ws + o; o += (size_t)NTOT * CDIM;      // hn / Z / hn2
  float* R2   = ws + o; o += (size_t)NTOT * 3 * CDIM;  // QKV / HI / Y
  float* R3   = ws + o; o += (size_t)NTOT * CHID;      // HS / YN / attn-out / F2
  float* R4   = ws + o; o += (size_t)NTOT * CHID;      // DT / F1
  float* T1   = ws + o; o += (size_t)NTOT * IRANK;     // router hidden
  float* POL  = ws + o; o += (size_t)NTOT * NEXP;      // logits -> policy
  float* PRM  = ws + o; o += (size_t)NTOT * DSTATE;    // prompt
  float* PSM  = ws + o; o += (size_t)NTOT * DSTATE;    // prompt sorted
  float* XD   = ws + o; o += (size_t)NTOT * LDXD;      // x_dbl (22 cols)
  float* FE   = ws + o; o += (size_t)NEXP * DSTATE;    // full_emb
  float* HEND = ws + o; o += (size_t)NB * CHID * DSTATE * NCH;  // scan chunk ends
  float* HINI = ws + o; o += (size_t)NB * CHID * DSTATE * NCH;  // scan chunk inits
  float* SDT  = ws + o; o += (size_t)NB * CHID * NCH;           // per-chunk sum(dt)
  int* KEY    = (int*)(ws + o); o += NTOT;
  int* ORD    = (int*)(ws + o); o += NTOT;
  int* INV    = (int*)(ws + o); o += NTOT;
  int* HIST   = (int*)(ws + o); o += NB * NEXP;

  ln_in_kernel<<<NTOT / 256, 256, 0, stream>>>(x, ln_w, ln_b, H);

  for (int i = 0; i < NDEPTH; ++i) {
    int j = i / 2;
    ln_rows_kernel<<<NTOT / 8, 256, 0, stream>>>(H, n1_w + i * CDIM, n1_b + i * CDIM, R1, CDIM);
    if ((i & 1) == 0) {
      // ----- ASSM block -----
      gemm_launch<CDIM, CDIM, IRANK, IRANK, IRANK, 1, false, 2>(
          stream, R1, rw1 + (size_t)j * CDIM * IRANK, rb1 + j * IRANK, T1, NTOT);
      gemm_launch<IRANK, IRANK, NEXP, NEXP, NEXP, 0, false, 2>(
          stream, T1, rw2 + (size_t)j * IRANK * NEXP, rb2 + j * NEXP, POL, NTOT);
      gemm_launch<IRANK, IRANK, DSTATE, DSTATE, DSTATE, 0, false, 1>(
          stream, embB + (size_t)j * NEXP * IRANK, tokenA, nullptr, FE, NEXP);
      hipMemsetAsync(HIST, 0, NB * NEXP * sizeof(int), stream);
      softmax_argmax_kernel<<<NTOT / 8, 256, 0, stream>>>(POL, KEY, HIST);
      sort_kernel<<<NB, NEXP, 0, stream>>>(KEY, HIST, ORD, INV);
      gemm_launch<NEXP, NEXP, DSTATE, DSTATE, DSTATE, 0, false, 2>(
          stream, POL, FE, nullptr, PRM, NTOT);
      gemm_launch<CDIM, CDIM, CHID, CHID, CHID, 0, false, 2>(
          stream, R1, in_w + (size_t)j * CDIM * CHID, in_b + j * CHID, R2, NTOT);
      gather_kernel<<<NTOT, 192, 0, stream>>>(R2, PRM, ORD, R3, PSM);
      gemm_launch<CHID, CHID, LDXD, LDXD, LDXD, 0, false, 2>(
          stream, R3, xproj_w + (size_t)j * CHID * LDXD, nullptr, XD, NTOT);
      gemm_launch<RKD, LDXD, CHID, CHID, CHID, 2, false, 2>(
          stream, XD, dt_w + (size_t)j * RKD * CHID, dt_b + j * CHID, R4, NTOT);
      const float* Alj = A_log + (size_t)j * CHID * DSTATE;
      scan_pass1<<<NB * (CHID / 16) * NCH, 256, 0, stream>>>(R3, R4, XD, Alj, HEND, SDT);
      scan_combine<<<NB * (CHID / 16), 256, 0, stream>>>(Alj, HEND, SDT, HINI);
      scan_pass2<<<NB * (CHID / 16) * NCH, 256, 0, stream>>>(R3, R4, XD, PSM, Alj,
                  Dskip + j * CHID, HINI, R2);
      ln_rows_kernel<<<NTOT / 8, 256, 0, stream>>>(R2, onorm_w + j * CHID,
                  onorm_b + j * CHID, R3, CHID);
      gemm_launch<CHID, CHID, CDIM, CDIM, CDIM, 0, false, 2>(
          stream, R3, out_w + (size_t)j * CHID * CDIM, out_b + j * CDIM, R1, NTOT);
      scatter_add_kernel<<<NTOT * CDIM / 256, 256, 0, stream>>>(R1, INV, H);
    } else {
      // ----- window attention -----
      gemm_launch<CDIM, CDIM, 3 * CDIM, 3 * CDIM, 3 * CDIM, 0, false, 2>(
          stream, R1, qkv_w + (size_t)j * CDIM * 3 * CDIM, qkv_b + (size_t)j * 3 * CDIM,
          R2, NTOT);
      winattn_kernel<<<(NB * (NSEQ / WINS) * NHEAD) / 8, 256, 0, stream>>>(R2, R3);
      gemm_launch<CDIM, CDIM, CDIM, CDIM, CDIM, 0, true, 2>(
          stream, R3, proj_w + (size_t)j * CDIM * CDIM, proj_b + j * CDIM, H, NTOT);
    }
    // ----- ConvFFN -----
    ln_rows_kernel<<<NTOT / 8, 256, 0, stream>>>(H, n2_w + i * CDIM, n2_b + i * CDIM, R1, CDIM);
    gemm_launch<CDIM, CDIM, CHID, CHID, CHID, 1, false, 2>(
        stream, R1, fc1_w + (size_t)i * CDIM * CHID, fc1_b + i * CHID, R4, NTOT);
    dwconv_kernel<<<(NTOT * CHID) / 256, 256, 0, stream>>>(R4, dw_w + (size_t)i * CHID * 125, R3);
    gemm_launch<CHID, CHID, CDIM, CDIM, CDIM, 0, true, 2>(
        stream, R3, fc2_w + (size_t)i * CHID * CDIM, fc2_b + i * CDIM, H, NTOT);
  }

  out_add_kernel<<<(NB * CDIM * NSEQ) / 256, 256, 0, stream>>>(H, x, (float*)d_out);
}